// CoordLSVotingWeighted_34445637714650
// MI455X (gfx1250) — compile-verified
//
#include <hip/hip_runtime.h>
#include <math.h>

typedef __attribute__((ext_vector_type(2))) float v2f;
typedef __attribute__((ext_vector_type(8))) float v8f;

namespace {
constexpr int   kB      = 8;
constexpr int   kH      = 128;
constexpr int   kW      = 128;
constexpr int   kNPix   = kH * kW;                      // 16384
constexpr int   kNC     = 9;                            // seg channels incl. background
constexpr int   kOC     = 8;                            // output classes (drop class 0)
constexpr int   kNP     = 9;                            // points
constexpr int   kNV     = 45;                           // 9 points * {R00,R01,R11,q0,q1}
constexpr int   kWaves  = 4;
constexpr int   kThreads= kWaves * 32;                  // 128 threads = 4 wave32
constexpr int   kIter   = 4;                            // 32-pixel groups per wave
constexpr int   kWgPerB = kNPix / (kWaves * kIter * 32);// 32 workgroups per batch
constexpr int   kAccPerWg = kOC * kNV;                  // 360 partial values per WG
constexpr float kBeta   = 1000000.0f;
}

// -------------------------------------------------------------------------
// Kernel 1: per-pixel math + per-(batch,class) masked reduction via WMMA.
// Per batch the reduction is the GEMM  V(45 x Npix) @ hot(Npix x 8).
// Each wave stages 32 pixels in LDS, then runs 8 K-steps (K=4) x 3 M-tiles
// of v_wmma_f32_16x16x4_f32 into 3 f32 accumulator tiles.
// -------------------------------------------------------------------------
__global__ __launch_bounds__(kThreads)
void clsv_accum(const float* __restrict__ seg,
                const float* __restrict__ dir,
                const float* __restrict__ wgt,
                float* __restrict__ partial)
{
    __shared__ float ldsV[kWaves][32][48];   // [wave][pixel-slot][value m]
    __shared__ float ldsH[kWaves][32][16];   // [wave][pixel-slot][class n]
    __shared__ float ldsC[kWaves][48][16];   // per-wave C tiles for reduction

    const int tid   = threadIdx.x;
    const int wave  = tid >> 5;
    const int lane  = tid & 31;
    const int b     = blockIdx.x / kWgPerB;
    const int wgInB = blockIdx.x % kWgPerB;

    const int col   = lane & 15;             // M-row (A) / N-col (B) within tile
    const int khalf = (lane >> 4) << 1;      // lanes 0-15 -> K{0,1}, 16-31 -> K{2,3}

    v8f c0 = {}; v8f c1 = {}; v8f c2 = {};

    for (int it = 0; it < kIter; ++it) {
        const int pix = wgInB * (kWaves * kIter * 32) + wave * (kIter * 32) + it * 32 + lane;
        const int hh  = pix >> 7;
        const int wwc = pix & 127;
        const float chh = ((float)hh  + 0.5f) * (1.0f / 128.0f);  // both axes / HEIGHT
        const float cww = ((float)wwc + 0.5f) * (1.0f / 128.0f);

        const size_t pbase = (size_t)b * kNPix + (size_t)pix;
        const float* sp = seg + pbase * kNC;
        const float* dp = dir + pbase * (kNP * 2);
        const float* wp = wgt + pbase * kNP;

        // near-argmax softmax over 9 channels, drop background (class 0)
        float s[kNC];
        #pragma unroll
        for (int i = 0; i < kNC; ++i) s[i] = sp[i];
        float mx = s[0];
        #pragma unroll
        for (int i = 1; i < kNC; ++i) mx = fmaxf(mx, s[i]);
        float e[kNC]; float den = 0.0f;
        #pragma unroll
        for (int i = 0; i < kNC; ++i) { e[i] = expf((s[i] - mx) * kBeta); den += e[i]; }
        const float rden = 1.0f / den;
        #pragma unroll
        for (int cc = 1; cc < kNC; ++cc) ldsH[wave][lane][cc - 1] = e[cc] * rden;
        #pragma unroll
        for (int cc = kOC; cc < 16; ++cc) ldsH[wave][lane][cc] = 0.0f;

        // per-point projection values
        #pragma unroll
        for (int p = 0; p < kNP; ++p) {
            const float wx = wp[p];
            // stable softplus: max(x,0) + log1p(exp(-|x|))
            const float sw = fmaxf(wx, 0.0f) + log1pf(expf(-fabsf(wx)));
            float nx = dp[2 * p], ny = dp[2 * p + 1];
            const float nrm = sqrtf(nx * nx + ny * ny);
            const float inv = (nrm == 0.0f) ? 0.0f : (1.0f / nrm); // divide_no_nan
            nx *= inv; ny *= inv;
            const float r00 = sw * (1.0f - nx * nx);
            const float r01 = sw * (-(nx * ny));
            const float r11 = sw * (1.0f - ny * ny);
            float* vrow = &ldsV[wave][lane][p * 5];
            vrow[0] = r00; vrow[1] = r01; vrow[2] = r11;
            vrow[3] = r00 * chh + r01 * cww;   // q0
            vrow[4] = r01 * chh + r11 * cww;   // q1
        }
        ldsV[wave][lane][45] = 0.0f;
        ldsV[wave][lane][46] = 0.0f;
        ldsV[wave][lane][47] = 0.0f;

        // 32 pixels -> 8 K-steps of 4; 3 M-tiles each
        #pragma unroll
        for (int kk = 0; kk < 8; ++kk) {
            const int kr = kk * 4 + khalf;
            v2f a0, a1, a2, bm;
            a0.x = ldsV[wave][kr    ][ 0 + col];
            a0.y = ldsV[wave][kr + 1][ 0 + col];
            a1.x = ldsV[wave][kr    ][16 + col];
            a1.y = ldsV[wave][kr + 1][16 + col];
            a2.x = ldsV[wave][kr    ][32 + col];
            a2.y = ldsV[wave][kr + 1][32 + col];
            bm.x = ldsH[wave][kr    ][col];
            bm.y = ldsH[wave][kr + 1][col];
            c0 = __builtin_amdgcn_wmma_f32_16x16x4_f32(false, a0, false, bm, (short)0, c0, false, false);
            c1 = __builtin_amdgcn_wmma_f32_16x16x4_f32(false, a1, false, bm, (short)0, c1, false, false);
            c2 = __builtin_amdgcn_wmma_f32_16x16x4_f32(false, a2, false, bm, (short)0, c2, false, false);
        }
    }

    // spill per-wave C tiles: lane l, vgpr r -> M = r + 8*(l>>4), N = l&15
    const int mhi = (lane >> 4) << 3;
    #pragma unroll
    for (int r = 0; r < 8; ++r) {
        ldsC[wave][ 0 + r + mhi][col] = c0[r];
        ldsC[wave][16 + r + mhi][col] = c1[r];
        ldsC[wave][32 + r + mhi][col] = c2[r];
    }
    __syncthreads();

    // deterministic fixed-order cross-wave reduction; one partial block per WG
    for (int idx = tid; idx < kAccPerWg; idx += kThreads) {
        const int cc = idx / kNV;
        const int m  = idx % kNV;
        float acc = 0.0f;
        #pragma unroll
        for (int wv = 0; wv < kWaves; ++wv) acc += ldsC[wv][m][cc];
        partial[(size_t)blockIdx.x * kAccPerWg + idx] = acc;
    }
}

// -------------------------------------------------------------------------
// Kernel 2: reduce the 32 per-WG partials per batch in f64 (fixed order),
// then closed-form pinv of the symmetric 2x2 (jnp rcond = 2*eps) and solve.
// -------------------------------------------------------------------------
__global__ void clsv_solve(const float* __restrict__ partial, float* __restrict__ out)
{
    const int idx = blockIdx.x * blockDim.x + threadIdx.x;
    if (idx >= kB * kOC * kNP) return;
    const int p  = idx % kNP;
    const int cc = (idx / kNP) % kOC;
    const int b  = idx / (kNP * kOC);

    double v[5] = {0.0, 0.0, 0.0, 0.0, 0.0};
    for (int wg = 0; wg < kWgPerB; ++wg) {
        const float* src = partial + ((size_t)(b * kWgPerB + wg)) * kAccPerWg + cc * kNV + p * 5;
        #pragma unroll
        for (int k = 0; k < 5; ++k) v[k] += (double)src[k];
    }
    const double A = v[0], Bv = v[1], Cc = v[2], q0 = v[3], q1 = v[4];

    // eigendecomposition of [[A,Bv],[Bv,Cc]]
    const double mm = 0.5 * (A + Cc);
    const double dd = 0.5 * (A - Cc);
    const double rr = sqrt(dd * dd + Bv * Bv);
    const double l1 = mm + rr;
    const double l2 = mm - rr;
    const double s1 = fabs(l1), s2 = fabs(l2);
    const double smax   = fmax(s1, s2);
    const double cutoff = 2.0 * 2.2204460492503131e-16 * smax;  // pinv rcond = max(M,N)*eps

    double v1x, v1y;
    if (Bv != 0.0)      { v1x = Bv;  v1y = l1 - A; }
    else if (A >= Cc)   { v1x = 1.0; v1y = 0.0;    }
    else                { v1x = 0.0; v1y = 1.0;    }
    const double nrm = sqrt(v1x * v1x + v1y * v1y);
    if (nrm > 0.0) { v1x /= nrm; v1y /= nrm; }
    const double v2x = -v1y, v2y = v1x;

    const double i1 = (s1 > cutoff) ? (1.0 / l1) : 0.0;
    const double i2 = (s2 > cutoff) ? (1.0 / l2) : 0.0;
    const double P00 = i1 * v1x * v1x + i2 * v2x * v2x;
    const double P01 = i1 * v1x * v1y + i2 * v2x * v2y;
    const double P11 = i1 * v1y * v1y + i2 * v2y * v2y;

    const double p0 = P00 * q0 + P01 * q1;
    const double p1 = P01 * q0 + P11 * q1;
    out[(size_t)idx * 2 + 0] = (float)(p0 * 128.0);   // * HEIGHT
    out[(size_t)idx * 2 + 1] = (float)(p1 * 128.0);
}

extern "C" void kernel_launch(void* const* d_in, const int* in_sizes, int n_in,
                              void* d_out, int out_size, void* d_ws, size_t ws_size,
                              hipStream_t stream)
{
    (void)in_sizes; (void)n_in; (void)out_size; (void)ws_size;
    const float* seg = (const float*)d_in[0];
    const float* dir = (const float*)d_in[1];
    const float* wgt = (const float*)d_in[2];
    float* partial = (float*)d_ws;   // kB*kWgPerB*kAccPerWg f32 = 368,640 bytes
    float* out     = (float*)d_out;

    clsv_accum<<<kB * kWgPerB, kThreads, 0, stream>>>(seg, dir, wgt, partial);

    const int n = kB * kOC * kNP;    // 576 solves
    clsv_solve<<<(n + 127) / 128, 128, 0, stream>>>(partial, out);
}